// Selective_CE_Dice_Loss_35107062677769
// MI455X (gfx1250) — compile-verified
//
#include <hip/hip_runtime.h>
#include <math.h>

#define BB 8
#define QQ 32
#define HWN 65536            // 256*256
#define SMOOTH 1e-4f

typedef __attribute__((ext_vector_type(2))) float v2f;
typedef __attribute__((ext_vector_type(8))) float v8f;

// Workspace (floats):
//  ws[0]                         : ce partial sum
//  ws[1 .. 1+B*Q)                : den[b][qp]   (psum + tsum, preds-channel space)
//  ws[1+B*Q .. 1+2*B*Q)          : inter[b][qp]
#define WS_CE    0
#define WS_DEN   1
#define WS_INT   (1 + BB*QQ)
#define WS_TOTAL (1 + 2*BB*QQ)   // 513 floats

__global__ void init_ws_kernel(float* __restrict__ ws) {
    int i = blockIdx.x * blockDim.x + threadIdx.x;
    if (i < WS_TOTAL) ws[i] = 0.0f;
}

__global__ __launch_bounds__(256) void selce_main_kernel(
        const float* __restrict__ preds,
        const int*   __restrict__ targets,
        const int*   __restrict__ object_num,
        float*       __restrict__ ws) {
    const int b    = blockIdx.y;
    const int numb = object_num[b];
    const float* pb = preds + (size_t)b * QQ * HWN;
    const int*   tb = targets + (size_t)b * HWN;

    // Register-resident per-lane accumulators (constant indices via full unroll).
    float accDen[QQ];   // psum + tsum contribution, preds-channel space
    float accInt[QQ];   // intersection contribution
    #pragma unroll
    for (int q = 0; q < QQ; ++q) { accDen[q] = 0.0f; accInt[q] = 0.0f; }
    float ceAcc = 0.0f;

    const int g      = blockIdx.x * blockDim.x + threadIdx.x;
    const int stride = gridDim.x * blockDim.x;

    for (int hw = g; hw < HWN; hw += stride) {
        const int  t    = tb[hw];
        const int  tp   = (t + QQ - 1) & (QQ - 1);       // preds-channel index of target
        const bool keep = (t != 0) && (t <= numb);

        float x[QQ];
        #pragma unroll
        for (int q = 0; q < QQ; ++q)                      // 32 coalesced b32 loads,
            x[q] = pb[(size_t)q * HWN + hw];              // single base + imm offsets

        float m  = x[0];
        float xt = x[0];
        #pragma unroll
        for (int q = 0; q < QQ; ++q) {
            m  = fmaxf(m, x[q]);
            xt = (q == tp) ? x[q] : xt;
        }

        float s = 0.0f;
        #pragma unroll
        for (int q = 0; q < QQ; ++q) {
            float e = __expf(x[q] - m);                   // v_exp_f32 path
            x[q] = e;
            s += e;
        }
        const float inv = __builtin_amdgcn_rcpf(s);

        #pragma unroll
        for (int q = 0; q < QQ; ++q) {
            const float p   = x[q] * inv;
            const float hit = (q == tp) ? 1.0f : 0.0f;
            accDen[q] += p + hit;                         // psum + tsum folded
            accInt[q] += hit * p;                         // intersection
        }
        ceAcc += keep ? (__logf(s) + m - xt) : 0.0f;      // -log p_t
    }

    // Block reduction: 65 LDS float atomics per lane, then 65 global atomics.
    __shared__ float lred[2 * QQ + 1];
    if (threadIdx.x < 2 * QQ + 1) lred[threadIdx.x] = 0.0f;
    __syncthreads();
    #pragma unroll
    for (int q = 0; q < QQ; ++q) {
        atomicAdd(&lred[q],      accDen[q]);
        atomicAdd(&lred[QQ + q], accInt[q]);
    }
    atomicAdd(&lred[2 * QQ], ceAcc);
    __syncthreads();

    const int tix = threadIdx.x;
    if (tix < QQ) {
        atomicAdd(&ws[WS_DEN + b * QQ + tix], lred[tix]);
    } else if (tix < 2 * QQ) {
        atomicAdd(&ws[WS_INT + b * QQ + (tix - QQ)], lred[tix]);
    } else if (tix == 2 * QQ) {
        atomicAdd(&ws[WS_CE], lred[2 * QQ]);
    }
}

// One wave32. Builds masked dice matrix (8 rows x 32 classes, scaled by 1/numb)
// as V_WMMA_F32_16X16X4_F32 A-fragments and row-reduces it against an all-ones
// B matrix, chaining C over 8 K-chunks. D[b][*] = dice_b[b] (replicated).
__global__ __launch_bounds__(32) void selce_finalize_kernel(
        const float* __restrict__ ws,
        const int*   __restrict__ object_num,
        float*       __restrict__ out) {
    const int lane  = threadIdx.x;           // 0..31
    const int m     = lane & 15;             // A-matrix row (b index; rows 8..15 = 0)
    const int kbase = (lane < 16) ? 0 : 2;   // 32-bit A 16x4: lanes>=16 hold K=2,3

    const int   numb = (m < BB) ? object_num[m] : 1;
    const float invn = 1.0f / (float)numb;

    v2f bones; bones.x = 1.0f; bones.y = 1.0f;   // all-ones B (layout-proof)
    v8f c = {};

    #pragma unroll
    for (int chunk = 0; chunk < 8; ++chunk) {
        v2f a;
        #pragma unroll
        for (int e = 0; e < 2; ++e) {
            const int q = chunk * 4 + kbase + e;      // inputs-space class 0..31
            float v = 0.0f;
            if (m < BB) {
                const int   qp    = (q + QQ - 1) & (QQ - 1);   // preds-channel remap
                const float den   = ws[WS_DEN + m * QQ + qp];
                const float inter = ws[WS_INT + m * QQ + qp];
                const float dice  = 1.0f - (2.0f * inter + SMOOTH) / (den + SMOOTH);
                const bool  valid = (q >= 1) && (q <= numb);
                v = valid ? dice * invn : 0.0f;
            }
            if (e == 0) a.x = v; else a.y = v;
        }
        // D = A(16x4,f32) * ones(4x16) + C  -> per-row partial sums
        c = __builtin_amdgcn_wmma_f32_16x16x4_f32(
                /*neg_a=*/false, a, /*neg_b=*/false, bones,
                /*c_mod=*/(short)0, c, /*reuse_a=*/false, /*reuse_b=*/false);
    }

    if (lane == 0) {
        // Rows 0..7 live in elements 0..7 of lanes 0..15 (any column).
        float dice_loss = 0.0f;
        #pragma unroll
        for (int b = 0; b < BB; ++b) dice_loss += c[b];
        dice_loss *= (1.0f / (float)BB);
        const float ce_loss = ws[WS_CE] / (float)(BB * HWN);
        out[0] = ce_loss + dice_loss;   // WEIGHT_BCE = WEIGHT_DC = 1
    }
}

extern "C" void kernel_launch(void* const* d_in, const int* in_sizes, int n_in,
                              void* d_out, int out_size, void* d_ws, size_t ws_size,
                              hipStream_t stream) {
    const float* preds      = (const float*)d_in[0];
    const int*   targets    = (const int*)d_in[1];
    const int*   object_num = (const int*)d_in[2];
    float*       out        = (float*)d_out;
    float*       ws         = (float*)d_ws;

    (void)in_sizes; (void)n_in; (void)out_size; (void)ws_size;

    init_ws_kernel<<<dim3(3), dim3(256), 0, stream>>>(ws);

    dim3 grid(64, BB);                 // 512 blocks x 8 waves, 4 pixels/lane
    selce_main_kernel<<<grid, dim3(256), 0, stream>>>(preds, targets, object_num, ws);

    selce_finalize_kernel<<<dim3(1), dim3(32), 0, stream>>>(ws, object_num, out);
}